// InterLevelMP_88751204204557
// MI455X (gfx1250) — compile-verified
//
#include <hip/hip_runtime.h>

// InterLevelMP for MI455X (gfx1250): fp32-exact WMMA (v_wmma_f32_16x16x4_f32)
// H = 128 throughout. wave32, wave-private LDS tiles, no cross-wave barriers.

typedef float v2f __attribute__((ext_vector_type(2)));
typedef float v8f __attribute__((ext_vector_type(8)));

constexpr int HD = 128;

static __device__ inline v8f splat8(float x) {
  v8f v = {x, x, x, x, x, x, x, x};
  return v;
}

// ---------------------------------------------------------------------------
// zero fill (graph-replay safe re-init of scratch)
__global__ __launch_bounds__(256) void zero_kernel(float* __restrict__ p, size_t n) {
  size_t i = (size_t)blockIdx.x * blockDim.x + threadIdx.x;
  size_t stride = (size_t)gridDim.x * blockDim.x;
  for (; i < n; i += stride) p[i] = 0.0f;
}

// degree counts (scatter-mean denominator)
__global__ __launch_bounds__(256) void count_kernel(const int* __restrict__ dst, int E,
                                                    float* __restrict__ cnt) {
  int i = blockIdx.x * blockDim.x + threadIdx.x;
  if (i < E) unsafeAtomicAdd(&cnt[dst[i]], 1.0f);
}

// ---------------------------------------------------------------------------
// Edge message: msgs = src_x[src] @ msg_w + msg_b, atomically scattered to agg.
// Block = 64 threads = 2 waves; each wave owns 32 edges (two 16-row M-tiles)
// so every B fragment load feeds TWO wmma ops (halves VMEM issue per FLOP).
// N is processed in two groups of 4 tiles to keep only 64 acc VGPRs live.
__global__ __launch_bounds__(64) void edge_msg_kernel(
    const float* __restrict__ srcX, const int* __restrict__ srcIdx,
    const int* __restrict__ dstIdx, int E,
    const float* __restrict__ W,     // (128,128) row-major [k][n]
    const float* __restrict__ bias,  // (128,)
    float* __restrict__ agg)         // (nDst,128)
{
  __shared__ float Xs[2][32][HD];  // 32 KB, one 32x128 slice per wave

  const int wave = threadIdx.x >> 5;
  const int lane = threadIdx.x & 31;
  const int base = (blockIdx.x * 2 + wave) * 32;
  if (base >= E) return;  // wave-uniform guard: EXEC stays all-ones for WMMA

  // gather 32 rows of src_x into this wave's LDS slice (float4, clamped tail)
  float4* xs4 = (float4*)&Xs[wave][0][0];
  const float4* gx4 = (const float4*)srcX;
  for (int i = lane; i < 32 * (HD / 4); i += 32) {
    int row = i >> 5;  // 32 float4 per row
    int q = i & 31;
    int e = base + row;
    int s = srcIdx[e < E ? e : (E - 1)];
    xs4[i] = gx4[(size_t)s * (HD / 4) + q];
  }
  // same-wave DS ops are in-order: stores above are visible to loads below.

  const int half = lane >> 4;   // 0: K%4 in {0,1}, 1: K%4 in {2,3}
  const int l16 = lane & 15;
  const int koff = half * 2;

  for (int g = 0; g < 2; ++g) {  // N-tile groups {0..3}, {4..7}
    v8f acc[2][4];
#pragma unroll
    for (int m = 0; m < 2; ++m)
#pragma unroll
      for (int t = 0; t < 4; ++t)
        acc[m][t] = splat8(bias[(g * 4 + t) * 16 + l16]);

    for (int kk = 0; kk < HD / 4; ++kk) {
      const int k0 = kk * 4 + koff;
      v2f a0, a1;
      a0.x = Xs[wave][l16][k0];
      a0.y = Xs[wave][l16][k0 + 1];
      a1.x = Xs[wave][16 + l16][k0];
      a1.y = Xs[wave][16 + l16][k0 + 1];
#pragma unroll
      for (int t = 0; t < 4; ++t) {
        const int n = (g * 4 + t) * 16 + l16;
        v2f b;
        b.x = W[k0 * HD + n];
        b.y = W[(k0 + 1) * HD + n];
        acc[0][t] = __builtin_amdgcn_wmma_f32_16x16x4_f32(false, a0, false, b, (short)0,
                                                          acc[0][t], false, false);
        acc[1][t] = __builtin_amdgcn_wmma_f32_16x16x4_f32(false, a1, false, b, (short)0,
                                                          acc[1][t], false, false);
      }
    }

    // C/D layout: v8f element r <-> row (r + 8*half), column lane%16 of n-tile.
#pragma unroll
    for (int m = 0; m < 2; ++m) {
#pragma unroll
      for (int r = 0; r < 8; ++r) {
        const int e = base + m * 16 + r + half * 8;
        if (e < E) {
          const size_t dn = (size_t)dstIdx[e] * HD;
#pragma unroll
          for (int t = 0; t < 4; ++t) {
            const int col = (g * 4 + t) * 16 + l16;
            unsafeAtomicAdd(&agg[dn + col], acc[m][t][r]);
          }
        }
      }
    }
  }
}

// ---------------------------------------------------------------------------
// Node update: gi=[x|agg/cnt]; gate=sigmoid(gi@Wg+bg); upd=relu(gi@W1+b1)@W2+b2;
// out = LN(gate*x + (1-gate)*upd).  Block = 64 threads = 2 waves, 16 nodes/wave.
// gate/u1 GEMMs interleave so each A fragment is shared by two wmma ops.
__global__ __launch_bounds__(64) void node_update_kernel(
    const float* __restrict__ dstX, const float* __restrict__ agg,
    const float* __restrict__ cnt, int nDst,
    const float* __restrict__ gateW, const float* __restrict__ gateB,
    const float* __restrict__ u1W, const float* __restrict__ u1B,
    const float* __restrict__ u2W, const float* __restrict__ u2B,
    const float* __restrict__ lnG, const float* __restrict__ lnB,
    float* __restrict__ out)
{
  __shared__ float gis[2][16][2 * HD];  // 32 KB: [x(0..127) | agg/cnt(128..255)]
  __shared__ float h1s[2][16][HD];      // 16 KB: relu(u1) tile, reused as y tile
  __shared__ float mstat[2][16];
  __shared__ float rstat[2][16];

  const int wave = threadIdx.x >> 5;
  const int lane = threadIdx.x & 31;
  const int tileBase = (blockIdx.x * 2 + wave) * 16;
  if (tileBase >= nDst) return;  // wave-uniform

  const int half = lane >> 4;
  const int l16 = lane & 15;
  const int koff = half * 2;

  // stage gi (float4): 16 rows x 64 float4 each (clamped for partial tiles)
  float4* g4 = (float4*)&gis[wave][0][0];
  const float4* x4 = (const float4*)dstX;
  const float4* a4 = (const float4*)agg;
  for (int i = lane; i < 16 * 64; i += 32) {
    int row = i >> 6;
    int q = i & 63;
    int node = tileBase + row;
    if (node >= nDst) node = nDst - 1;
    float4 v;
    if (q < 32) {
      v = x4[(size_t)node * 32 + q];
    } else {
      float inv = 1.0f / fmaxf(cnt[node], 1.0f);
      v = a4[(size_t)node * 32 + (q - 32)];
      v.x *= inv; v.y *= inv; v.z *= inv; v.w *= inv;
    }
    g4[i] = v;
  }

  // gate and u1 GEMMs over K = 256 (interleaved so A-frag is shared)
  v8f gacc[8], uacc[8];
  for (int t = 0; t < 8; ++t) {
    gacc[t] = splat8(gateB[t * 16 + l16]);
    uacc[t] = splat8(u1B[t * 16 + l16]);
  }
  for (int kk = 0; kk < (2 * HD) / 4; ++kk) {
    const int k0 = kk * 4 + koff;
    v2f a;
    a.x = gis[wave][l16][k0];
    a.y = gis[wave][l16][k0 + 1];
#pragma unroll
    for (int t = 0; t < 8; ++t) {
      const int n = t * 16 + l16;
      v2f b;
      b.x = gateW[k0 * HD + n];
      b.y = gateW[(k0 + 1) * HD + n];
      gacc[t] = __builtin_amdgcn_wmma_f32_16x16x4_f32(false, a, false, b, (short)0,
                                                      gacc[t], false, false);
      v2f b2;
      b2.x = u1W[k0 * HD + n];
      b2.y = u1W[(k0 + 1) * HD + n];
      uacc[t] = __builtin_amdgcn_wmma_f32_16x16x4_f32(false, a, false, b2, (short)0,
                                                      uacc[t], false, false);
    }
  }

  // relu(u1) -> LDS (A-operand layout for u2 GEMM)
#pragma unroll
  for (int t = 0; t < 8; ++t) {
    const int col = t * 16 + l16;
#pragma unroll
    for (int r = 0; r < 8; ++r) {
      h1s[wave][r + half * 8][col] = fmaxf(uacc[t][r], 0.0f);
    }
  }

  // u2 GEMM over K = 128, reuse uacc as accumulator
  for (int t = 0; t < 8; ++t) uacc[t] = splat8(u2B[t * 16 + l16]);
  for (int kk = 0; kk < HD / 4; ++kk) {
    const int k0 = kk * 4 + koff;
    v2f a;
    a.x = h1s[wave][l16][k0];
    a.y = h1s[wave][l16][k0 + 1];
#pragma unroll
    for (int t = 0; t < 8; ++t) {
      const int n = t * 16 + l16;
      v2f b;
      b.x = u2W[k0 * HD + n];
      b.y = u2W[(k0 + 1) * HD + n];
      uacc[t] = __builtin_amdgcn_wmma_f32_16x16x4_f32(false, a, false, b, (short)0,
                                                      uacc[t], false, false);
    }
  }

  // y = gate*x + (1-gate)*upd  (h1s reused as y tile; u2 reads are done)
#pragma unroll
  for (int t = 0; t < 8; ++t) {
    const int col = t * 16 + l16;
#pragma unroll
    for (int r = 0; r < 8; ++r) {
      const int row = r + half * 8;
      float x = gis[wave][row][col];
      float g = 1.0f / (1.0f + expf(-gacc[t][r]));
      h1s[wave][row][col] = g * x + (1.0f - g) * uacc[t][r];
    }
  }

  // LayerNorm: lanes 0..15 compute per-row mean / rstd
  if (lane < 16) {
    const float4* y4 = (const float4*)&h1s[wave][lane][0];
    float s = 0.0f, s2 = 0.0f;
    for (int q = 0; q < HD / 4; ++q) {
      float4 v = y4[q];
      s += v.x + v.y + v.z + v.w;
      s2 += v.x * v.x + v.y * v.y + v.z * v.z + v.w * v.w;
    }
    float m = s * (1.0f / HD);
    float var = s2 * (1.0f / HD) - m * m;
    mstat[wave][lane] = m;
    rstat[wave][lane] = rsqrtf(var + 1e-5f);
  }

  // normalize + coalesced store (masked for partial tiles)
  for (int idx = lane; idx < 16 * HD; idx += 32) {
    int row = idx >> 7;
    int col = idx & (HD - 1);
    if (tileBase + row < nDst) {
      float y = h1s[wave][row][col];
      float o = (y - mstat[wave][row]) * rstat[wave][row] * lnG[col] + lnB[col];
      out[(size_t)(tileBase + row) * HD + col] = o;
    }
  }
}

// ---------------------------------------------------------------------------
static void run_direction(const float* srcX, const float* dstX,
                          const int* srcIdx, const int* dstIdx, int E, int nDst,
                          const float* msgW, const float* msgB,
                          const float* gateW, const float* gateB,
                          const float* u1W, const float* u1B,
                          const float* u2W, const float* u2B,
                          const float* lnG, const float* lnB,
                          float* agg, float* cnt, float* out, hipStream_t stream) {
  zero_kernel<<<2048, 256, 0, stream>>>(agg, (size_t)nDst * HD);
  zero_kernel<<<256, 256, 0, stream>>>(cnt, (size_t)nDst);
  count_kernel<<<(E + 255) / 256, 256, 0, stream>>>(dstIdx, E, cnt);
  edge_msg_kernel<<<(E + 63) / 64, 64, 0, stream>>>(srcX, srcIdx, dstIdx, E,
                                                    msgW, msgB, agg);
  int tiles = (nDst + 15) / 16;
  node_update_kernel<<<(tiles + 1) / 2, 64, 0, stream>>>(
      dstX, agg, cnt, nDst, gateW, gateB, u1W, u1B, u2W, u2B, lnG, lnB, out);
}

extern "C" void kernel_launch(void* const* d_in, const int* in_sizes, int n_in,
                              void* d_out, int out_size, void* d_ws, size_t ws_size,
                              hipStream_t stream) {
  const float* lowerX = (const float*)d_in[0];
  const float* upperX = (const float*)d_in[1];
  const int* buE = (const int*)d_in[2];
  const int* tdE = (const int*)d_in[3];

  const float* bu_msg_w = (const float*)d_in[4];
  const float* bu_msg_b = (const float*)d_in[5];
  const float* bu_gate_w = (const float*)d_in[6];
  const float* bu_gate_b = (const float*)d_in[7];
  const float* bu_u1_w = (const float*)d_in[8];
  const float* bu_u1_b = (const float*)d_in[9];
  const float* bu_u2_w = (const float*)d_in[10];
  const float* bu_u2_b = (const float*)d_in[11];
  const float* bu_ln_g = (const float*)d_in[12];
  const float* bu_ln_b = (const float*)d_in[13];

  const float* td_msg_w = (const float*)d_in[14];
  const float* td_msg_b = (const float*)d_in[15];
  const float* td_gate_w = (const float*)d_in[16];
  const float* td_gate_b = (const float*)d_in[17];
  const float* td_u1_w = (const float*)d_in[18];
  const float* td_u1_b = (const float*)d_in[19];
  const float* td_u2_w = (const float*)d_in[20];
  const float* td_u2_b = (const float*)d_in[21];
  const float* td_ln_g = (const float*)d_in[22];
  const float* td_ln_b = (const float*)d_in[23];

  const int nLower = in_sizes[0] / HD;
  const int nUpper = in_sizes[1] / HD;
  const int E = in_sizes[2] / 2;

  float* out = (float*)d_out;
  float* updLower = out;                              // output order: (lower, upper)
  float* updUpper = out + (size_t)nLower * HD;

  // shared scratch: agg sized for the larger direction, cnt right after
  float* agg = (float*)d_ws;
  float* cnt = agg + (size_t)nLower * HD;

  // bottom-up: lower -> upper
  run_direction(lowerX, upperX, buE, buE + E, E, nUpper,
                bu_msg_w, bu_msg_b, bu_gate_w, bu_gate_b,
                bu_u1_w, bu_u1_b, bu_u2_w, bu_u2_b, bu_ln_g, bu_ln_b,
                agg, cnt, updUpper, stream);

  // top-down: upper -> lower
  run_direction(upperX, lowerX, tdE, tdE + E, E, nLower,
                td_msg_w, td_msg_b, td_gate_w, td_gate_b,
                td_u1_w, td_u1_b, td_u2_w, td_u2_b, td_ln_g, td_ln_b,
                agg, cnt, updLower, stream);
}